// EncoderLSTM_26620207300795
// MI455X (gfx1250) — compile-verified
//
#include <hip/hip_runtime.h>
#include <math.h>
#include <stdint.h>

// ---------------------------------------------------------------------------
// Bidirectional LSTM encoder for MI455X (gfx1250, wave32).
//
//  Kernel P: pack U into WMMA B-fragment order (one-time, 2 MB).
//  Kernel 1: xz[d][btile][t][16][1024] = x @ W + b   (fp32 WMMA 16x16x4,
//            LDS-staged W, tile-major output, NT stores so the 537 MB xz
//            stream does not evict U/x/W from the 192 MB L2).
//  Kernel 2: persistent per-(direction, batch-tile-of-16) scan. Batch rows are
//            independent -> no cross-WG sync. Per step: next xz tile is
//            async-prefetched into LDS (GLOBAL_LOAD_ASYNC_TO_LDS_B128 +
//            s_wait_asynccnt), z = xz + h@U via WMMA f32 (h in LDS, U
//            fragments coalesced from L2), gates in VALU, c-state in regs,
//            hs written with NT stores (write-once stream).
// ---------------------------------------------------------------------------

typedef __attribute__((ext_vector_type(2))) float v2f;
typedef __attribute__((ext_vector_type(4))) float v4f;
typedef __attribute__((ext_vector_type(8))) float v8f;

#define T_LEN   1024
#define BATCH   64
#define FEAT_N  512
#define UNITS_N 256
#define NG      1024            // 4 * UNITS
#define ENC_SZ  (BATCH * T_LEN * 512)

// Flip to 0 if the async-to-LDS mnemonic is rejected -> synchronous LDS copy.
#define USE_ASYNC_TILE 1

static __device__ __forceinline__ v8f wmma_f32_16x16x4(v2f a, v2f b, v8f c) {
  // 8 args: (neg_a, A, neg_b, B, c_mod, C, reuse_a, reuse_b)
  return __builtin_amdgcn_wmma_f32_16x16x4_f32(false, a, false, b, (short)0, c,
                                               false, false);
}

static __device__ __forceinline__ float sigmoidf_(float x) {
  return 1.0f / (1.0f + __expf(-x));
}

// ---------------------------------------------------------------------------
// Kernel P: pack U[256][1024] into fragment order UP[ntile][kt][lane][2]:
//   UP element (n, kt, lane, j) = U[4*kt + 2*(lane>>4) + j][16*n + (lane&15)]
// so kernel 2 loads one aligned, fully-coalesced b64 per lane per k-slab.
// ---------------------------------------------------------------------------
__global__ __launch_bounds__(256)
void lstm_pack_u(const float* __restrict__ Uf, const float* __restrict__ Ub,
                 float* __restrict__ UP)
{
  const int d = blockIdx.y;
  const float* __restrict__ U = d ? Ub : Uf;
  const int idx = blockIdx.x * 256 + threadIdx.x;    // 0 .. 262143
  const int j   = idx & 1;
  const int ln  = (idx >> 1) & 31;
  const int kt  = (idx >> 6) & 63;
  const int n   = idx >> 12;
  const int krow = 4 * kt + 2 * (ln >> 4) + j;
  const int col  = 16 * n + (ln & 15);
  UP[(size_t)d * (64 * 64 * 64) + idx] = U[(size_t)krow * NG + col];
}

// ---------------------------------------------------------------------------
// Kernel 1: input projection GEMM.
// grid = (BATCH*T/128, NG/64, 2 dirs), block = 256 (8 waves).
// ---------------------------------------------------------------------------
__global__ __launch_bounds__(256)
void lstm_xw_gemm(const float* __restrict__ X,
                  const float* __restrict__ Wf, const float* __restrict__ bf,
                  const float* __restrict__ Wb, const float* __restrict__ bb,
                  float* __restrict__ XZ)
{
  const int d = blockIdx.z;
  const float* __restrict__ W  = d ? Wb : Wf;
  const float* __restrict__ bv = d ? bb : bf;
  const int mbase = blockIdx.x * 128;
  const int nbase = blockIdx.y * 64;

  __shared__ float wt[64 * 132];   // W^T chunk [col 0..63][k 0..127]

  const int tid  = threadIdx.x;
  const int wave = tid >> 5;
  const int lane = tid & 31;
  const int lm   = lane & 15;
  const int lg   = lane >> 4;

  const int rI   = mbase + wave * 16 + lm;      // A-fragment row
  const int b    = rI >> 10;
  const int t    = rI & 1023;                   // scan-order time index
  const int tsrc = d ? (T_LEN - 1 - t) : t;     // bwd reads reversed x
  const float* __restrict__ xrow = X + (size_t)(b * T_LEN + tsrc) * FEAT_N;
  __builtin_prefetch(xrow, 0, 0);               // global_prefetch_b8

  v8f acc[4] = {};

  for (int kc = 0; kc < 4; ++kc) {              // K chunks of 128
    __syncthreads();
    for (int i = tid; i < 128 * 64; i += 256) {
      const int k = i >> 6;
      const int c = i & 63;
      wt[c * 132 + k] = W[(size_t)(kc * 128 + k) * NG + nbase + c];
    }
    __syncthreads();

    const float* xk = xrow + kc * 128;
    #pragma unroll 4
    for (int kt = 0; kt < 32; ++kt) {
      const int k0 = 4 * kt + 2 * lg;
      v2f a = *(const v2f*)(xk + k0);
      #pragma unroll
      for (int j = 0; j < 4; ++j) {
        v2f bfrag = *(const v2f*)(wt + (16 * j + lm) * 132 + k0);
        acc[j] = wmma_f32_16x16x4(a, bfrag, acc[j]);
      }
    }
  }

  // Bias + NT store into tile-major layout [d][btile][t][row-in-tile][1024].
  #pragma unroll
  for (int j = 0; j < 4; ++j) {
    const int col = nbase + 16 * j + lm;
    const float bias = bv[col];
    #pragma unroll
    for (int r = 0; r < 8; ++r) {
      const int rowI = mbase + wave * 16 + r + 8 * lg;
      const int bb2  = rowI >> 10;
      const int tt   = rowI & 1023;
      const int bt   = bb2 >> 4;
      const int rl   = bb2 & 15;
      float* dst = XZ + ((size_t)((d * 4 + bt) * T_LEN + tt) * 16 + rl) * NG + col;
      __builtin_nontemporal_store(acc[j][r] + bias, dst);
    }
  }
}

// ---------------------------------------------------------------------------
// Kernel 2: recurrent scan.
// ---------------------------------------------------------------------------
#define H_STRIDE  260    // 260 % 64 == 4 -> conflict-free A-fragment reads
#define Z_STRIDE  1026   // lane-group bank-disjoint exchange
#define XT_STRIDE 1028   // tile buffer stride (16B-aligned rows, no conflicts)

static __device__ __forceinline__ void issue_tile(const float* __restrict__ g,
                                                  float* l, int tid)
{
  // Copy one 16x1024 f32 tile (contiguous in global) into padded LDS.
  // Each of the 1024 threads moves 4 x b128.
  #pragma unroll
  for (int q = 0; q < 4; ++q) {
    const int linear = q * 4096 + tid * 4;      // float index
    const int row = linear >> 10;
    const int col = linear & 1023;
    const float* gs = g + linear;
    float* ls = l + row * XT_STRIDE + col;
#if USE_ASYNC_TILE
    const uint32_t laddr = (uint32_t)(uintptr_t)ls;  // LDS byte offset
    asm volatile("global_load_async_to_lds_b128 %0, %1, off"
                 :: "v"(laddr), "v"(gs) : "memory");
#else
    *(v4f*)ls = *(const v4f*)gs;
#endif
  }
}

__global__ __launch_bounds__(1024)
void lstm_recur(const float* __restrict__ UP, const float* __restrict__ XZ,
                float* __restrict__ out)
{
  const int btile = blockIdx.x;   // 0..3 (16 batch rows each)
  const int d     = blockIdx.y;   // 0 = fwd, 1 = bwd

  __shared__ float h_lds[16 * H_STRIDE];
  __shared__ float z_lds[16 * Z_STRIDE];
  __shared__ float xzb[2][16 * XT_STRIDE];   // double-buffered xz tile

  const int tid  = threadIdx.x;
  const int wave = tid >> 5;
  const int lane = tid & 31;
  const int lm   = lane & 15;
  const int lg   = lane >> 4;
  const int cb   = 32 * wave;                // this wave's first z column

  const float* __restrict__ xzg =
      XZ + (size_t)(d * 4 + btile) * T_LEN * (16 * NG);
  const float* __restrict__ up0 =
      UP + (size_t)d * (64 * 64 * 64) + (size_t)(2 * wave) * (64 * 64) + lane * 2;
  const float* __restrict__ up1 = up0 + 64 * 64;

  for (int i = tid; i < 16 * H_STRIDE; i += 1024) h_lds[i] = 0.0f;

  // Hoisted LDS bases (constant over the scan).
  const float* ha = h_lds + lm * H_STRIDE + 2 * lg;          // A-frag base
  float* zs       = z_lds + (8 * lg) * Z_STRIDE + cb + lm;   // z store base
  const int unit  = 16 * (wave & 15) + lm;                   // gate waves only
  const float* zg = z_lds + (8 * lg) * Z_STRIDE + unit;
  float* hw       = h_lds + (8 * lg) * H_STRIDE + unit;
  const int bg0   = btile * 16 + 8 * lg;
  float* outp  = out + ((size_t)bg0 * T_LEN + (size_t)(d ? T_LEN - 1 : 0)) * 512
                     + d * UNITS_N + unit;
  float* outp4 = outp + (size_t)4 * T_LEN * 512;
  const ptrdiff_t tstep = d ? -512 : 512;

  float creg[8];
  #pragma unroll
  for (int r = 0; r < 8; ++r) creg[r] = 0.0f;

  issue_tile(xzg, xzb[0], tid);              // prefetch tile t=0

  for (int t = 0; t < T_LEN; ++t) {
#if USE_ASYNC_TILE
    asm volatile("s_wait_asynccnt 0x0" ::: "memory");
#endif
    __syncthreads();   // h ready + current tile visible to all waves

    // Seed z accumulators from the LDS-resident xz tile.
    const float* xt = xzb[t & 1] + (8 * lg) * XT_STRIDE + cb + lm;
    v8f z0 = {}, z1 = {};
    #pragma unroll
    for (int r = 0; r < 8; ++r) {
      z0[r] = xt[r * XT_STRIDE];
      z1[r] = xt[r * XT_STRIDE + 16];
    }

    if (t + 1 < T_LEN)                        // prefetch next tile
      issue_tile(xzg + (size_t)(t + 1) * (16 * NG), xzb[(t + 1) & 1], tid);

    // z += h @ U : M=16, N=32 (2 tiles), K=256 (64 slabs of 4)
    #pragma unroll 4
    for (int kt = 0; kt < 64; ++kt) {
      v2f a  = *(const v2f*)(ha  + kt * 4);    // ds_load_b64, all-bank
      v2f bA = *(const v2f*)(up0 + kt * 64);   // coalesced 256B b64
      v2f bB = *(const v2f*)(up1 + kt * 64);
      z0 = wmma_f32_16x16x4(a, bA, z0);
      z1 = wmma_f32_16x16x4(a, bB, z1);
    }

    #pragma unroll
    for (int r = 0; r < 8; ++r) {
      zs[r * Z_STRIDE]      = z0[r];
      zs[r * Z_STRIDE + 16] = z1[r];
    }
    __syncthreads();   // z complete

    if (wave < 16) {   // gate waves: unit = 16*wave + lm, rows r + 8*lg
      #pragma unroll
      for (int r = 0; r < 8; ++r) {
        const float zi  = zg[r * Z_STRIDE];
        const float zf  = zg[r * Z_STRIDE + UNITS_N];
        const float zgt = zg[r * Z_STRIDE + 2 * UNITS_N];
        const float zo  = zg[r * Z_STRIDE + 3 * UNITS_N];
        const float ig = sigmoidf_(zi);
        const float fg = sigmoidf_(zf);
        const float gg = tanhf(zgt);
        const float og = sigmoidf_(zo);
        const float cn = fg * creg[r] + ig * gg;
        creg[r] = cn;
        const float hn = og * tanhf(cn);
        hw[r * H_STRIDE] = hn;
        if (r < 4)
          __builtin_nontemporal_store(hn, &outp[(size_t)r * T_LEN * 512]);
        else
          __builtin_nontemporal_store(hn, &outp4[(size_t)(r - 4) * T_LEN * 512]);
        if (t == T_LEN - 1) {                 // h_last / c_last
          const int bg = bg0 + r;
          out[(size_t)ENC_SZ + bg * 512 + d * UNITS_N + unit] = hn;
          out[(size_t)ENC_SZ + BATCH * 512 + bg * 512 + d * UNITS_N + unit] = cn;
        }
      }
      outp += tstep;
      outp4 += tstep;
    }
  }
}

// ---------------------------------------------------------------------------
extern "C" void kernel_launch(void* const* d_in, const int* in_sizes, int n_in,
                              void* d_out, int out_size, void* d_ws,
                              size_t ws_size, hipStream_t stream)
{
  (void)in_sizes; (void)n_in; (void)out_size; (void)ws_size;

  const float* x  = (const float*)d_in[0];
  const float* Wf = (const float*)d_in[1];
  const float* Uf = (const float*)d_in[2];
  const float* bf = (const float*)d_in[3];
  const float* Wb = (const float*)d_in[4];
  const float* Ub = (const float*)d_in[5];
  const float* bb = (const float*)d_in[6];
  float* out = (float*)d_out;

  float* XZ = (float*)d_ws;                         // 2*64*1024*1024 f32 = 512 MiB
  float* UP = XZ + (size_t)2 * BATCH * T_LEN * NG;  // + 2 MiB packed U

  lstm_pack_u<<<dim3(1024, 2), dim3(256), 0, stream>>>(Uf, Ub, UP);
  lstm_xw_gemm<<<dim3(BATCH * T_LEN / 128, NG / 64, 2), dim3(256), 0, stream>>>(
      x, Wf, bf, Wb, bb, XZ);
  lstm_recur<<<dim3(4, 2), dim3(1024), 0, stream>>>(UP, XZ, out);
}